// AttentionDecoder_2508260901147
// MI455X (gfx1250) — compile-verified
//
#include <hip/hip_runtime.h>
#include <hip/hip_bf16.h>
#include <math.h>

// Problem constants (from reference)
#define B_  32
#define L_  128
#define T_  127
#define S_  256
#define H_  512
#define G_  2048   // 4*H
#define V_  8000
#define NH_ 8
#define HD_ 64

#define LDA_  36    // padded LDS row stride (floats): 144B -> 16B aligned, bank-spread
#define KC_   32    // K staged per chunk

typedef __attribute__((ext_vector_type(2))) float v2f;
typedef __attribute__((ext_vector_type(8))) float v8f;

#if defined(__HIP_DEVICE_COMPILE__)
#if !__has_builtin(__builtin_amdgcn_wmma_f32_16x16x4_f32)
#error "device pass: missing __builtin_amdgcn_wmma_f32_16x16x4_f32"
#endif
#if __has_builtin(__builtin_amdgcn_global_load_async_to_lds_b128) && \
    __has_builtin(__builtin_amdgcn_s_wait_asynccnt)
#define USE_ASYNC_LDS 1
#else
#define USE_ASYNC_LDS 0
#endif
#else
#define USE_ASYNC_LDS 0
#endif

// Pointer-to-int4 type matching the builtin's diagnosed parameter type:
//   '__attribute__((__vector_size__(4 * sizeof(int)))) int *'
typedef int v4i_ld __attribute__((vector_size(16)));

__device__ __forceinline__ v8f wmma4(v2f a, v2f b, v8f c) {
  // (neg_a, A, neg_b, B, c_mod, C, reuse_a, reuse_b)
  return __builtin_amdgcn_wmma_f32_16x16x4_f32(false, a, false, b, (short)0, c,
                                               false, false);
}

__device__ __forceinline__ float sigmoidf_(float x) {
  return 1.0f / (1.0f + __expf(-x));
}

// 8 k-steps of 4 from LDS chunk; Ab/Wb already offset by (row*LDA_ + koff).
__device__ __forceinline__ void compute_chunk(const float* Ab, const float* Wb,
                                              v8f& c0, v8f& c1, v8f& c2, v8f& c3) {
#pragma unroll
  for (int kk = 0; kk < KC_; kk += 4) {
    v2f a  = *(const v2f*)(Ab + kk);
    v2f b0 = *(const v2f*)(Wb + kk);
    v2f b1 = *(const v2f*)(Wb + 16 * LDA_ + kk);
    v2f b2 = *(const v2f*)(Wb + 32 * LDA_ + kk);
    v2f b3 = *(const v2f*)(Wb + 48 * LDA_ + kk);
    c0 = wmma4(a, b0, c0);
    c1 = wmma4(a, b1, c1);
    c2 = wmma4(a, b2, c2);
    c3 = wmma4(a, b3, c3);
  }
}

// ---------------------------------------------------------------------------
// LDS double-buffered WMMA GEMM:  C[M,N] = A[M,K] @ W[N,K]^T (+bias) (+res)
// Block = 256 threads = 8 waves (4 m-groups x 2 n-groups), macro-tile 64x128.
// GATHER variant replaces A rows with embedding[targets[row]].
// K must be a multiple of 32 (here always 512).
// ---------------------------------------------------------------------------
template <bool GATHER>
__global__ __launch_bounds__(256) void gemm_tiled(
    const int* __restrict__ targets, const float* __restrict__ emb,
    const float* __restrict__ A, const float* __restrict__ W,
    const float* __restrict__ bias, const float* __restrict__ res,
    long res_stride, float* __restrict__ C, int M, int N, int K, int nb128) {
  __shared__ float As[2][64 * LDA_];    // 18 KB
  __shared__ float Ws[2][128 * LDA_];   // 36 KB

  const int tid = threadIdx.x;
  const int mb = blockIdx.x / nb128;
  const int nb = blockIdx.x % nb128;
  const int mBase = mb * 64, nBase = nb * 128;

  // ---- cooperative load mapping: 2 A-float4 + 4 W-float4 per thread ----
  const float* aSrc[2];
  int aLds[2];
#pragma unroll
  for (int u = 0; u < 2; ++u) {
    const int idx = tid + 256 * u;        // 0..511
    const int row = idx >> 3;             // 0..63
    const int col = (idx & 7) << 2;       // 0,4,..,28
    int grow = mBase + row;
    if (grow > M - 1) grow = M - 1;       // clamp (stores are guarded)
    const float* base;
    if (GATHER) {
      const int bb = grow / T_, tt = grow % T_;
      base = emb + (size_t)targets[bb * L_ + tt] * (size_t)K;
    } else {
      base = A + (size_t)grow * K;
    }
    aSrc[u] = base + col;
    aLds[u] = row * LDA_ + col;
  }
  const float* wSrc[4];
  int wLds[4];
#pragma unroll
  for (int u = 0; u < 4; ++u) {
    const int idx = tid + 256 * u;        // 0..1023
    const int row = idx >> 3;             // 0..127
    const int col = (idx & 7) << 2;
    int grow = nBase + row;
    if (grow > N - 1) grow = N - 1;
    wSrc[u] = W + (size_t)grow * K + col;
    wLds[u] = row * LDA_ + col;
  }

  const int lane = tid & 31;
  const int wid = tid >> 5;
  const int wm = wid & 3;                 // m-group 0..3 (16 rows each)
  const int wn = wid >> 2;                // n-group 0..1 (64 cols each)
  const int lr = lane & 15;
  const int koff = (lane >> 4) << 1;      // half-wave K pair select

  v8f c0 = {}, c1 = {}, c2 = {}, c3 = {};
  const int nChunks = K >> 5;             // K / 32

#if USE_ASYNC_LDS
  // ---- CDNA5 async global->LDS staging (ASYNCcnt) ----
#pragma unroll
  for (int u = 0; u < 2; ++u)
    __builtin_amdgcn_global_load_async_to_lds_b128(
        (v4i_ld*)(void*)aSrc[u], (v4i_ld*)(void*)&As[0][aLds[u]], 0, 0);
#pragma unroll
  for (int u = 0; u < 4; ++u)
    __builtin_amdgcn_global_load_async_to_lds_b128(
        (v4i_ld*)(void*)wSrc[u], (v4i_ld*)(void*)&Ws[0][wLds[u]], 0, 0);
  __builtin_amdgcn_s_wait_asynccnt(0);
  __syncthreads();

  for (int c = 0; c < nChunks; ++c) {
    const int buf = c & 1;
    if (c + 1 < nChunks) {
      const int k0 = (c + 1) << 5;
#pragma unroll
      for (int u = 0; u < 2; ++u)
        __builtin_amdgcn_global_load_async_to_lds_b128(
            (v4i_ld*)(void*)(aSrc[u] + k0),
            (v4i_ld*)(void*)&As[buf ^ 1][aLds[u]], 0, 0);
#pragma unroll
      for (int u = 0; u < 4; ++u)
        __builtin_amdgcn_global_load_async_to_lds_b128(
            (v4i_ld*)(void*)(wSrc[u] + k0),
            (v4i_ld*)(void*)&Ws[buf ^ 1][wLds[u]], 0, 0);
    }
    compute_chunk(&As[buf][(wm * 16 + lr) * LDA_ + koff],
                  &Ws[buf][(wn * 64 + lr) * LDA_ + koff], c0, c1, c2, c3);
    if (c + 1 < nChunks) {
      __builtin_amdgcn_s_wait_asynccnt(0);
      __syncthreads();
    }
  }
#else
  // ---- fallback: register-staged global_load_b128 -> ds_store_b128 ----
  float4 aReg[2], wReg[4];
#pragma unroll
  for (int u = 0; u < 2; ++u) aReg[u] = *(const float4*)aSrc[u];
#pragma unroll
  for (int u = 0; u < 4; ++u) wReg[u] = *(const float4*)wSrc[u];
#pragma unroll
  for (int u = 0; u < 2; ++u) *(float4*)&As[0][aLds[u]] = aReg[u];
#pragma unroll
  for (int u = 0; u < 4; ++u) *(float4*)&Ws[0][wLds[u]] = wReg[u];
  __syncthreads();

  for (int c = 0; c < nChunks; ++c) {
    const int buf = c & 1;
    if (c + 1 < nChunks) {
      const int k0 = (c + 1) << 5;        // issue next chunk's global loads now
#pragma unroll
      for (int u = 0; u < 2; ++u) aReg[u] = *(const float4*)(aSrc[u] + k0);
#pragma unroll
      for (int u = 0; u < 4; ++u) wReg[u] = *(const float4*)(wSrc[u] + k0);
    }
    compute_chunk(&As[buf][(wm * 16 + lr) * LDA_ + koff],
                  &Ws[buf][(wn * 64 + lr) * LDA_ + koff], c0, c1, c2, c3);
    if (c + 1 < nChunks) {
      __syncthreads();                    // everyone done reading buf^1
#pragma unroll
      for (int u = 0; u < 2; ++u) *(float4*)&As[buf ^ 1][aLds[u]] = aReg[u];
#pragma unroll
      for (int u = 0; u < 4; ++u) *(float4*)&Ws[buf ^ 1][wLds[u]] = wReg[u];
      __syncthreads();
    }
  }
#endif

  // ---- epilogue ----
  const int rbase = mBase + wm * 16 + ((lane >> 4) << 3);
  const int cbase = nBase + wn * 64;
#pragma unroll
  for (int i = 0; i < 8; ++i) {
    const int row = rbase + i;
    if (row < M) {
      const size_t ro = (size_t)row * N;
#pragma unroll
      for (int j = 0; j < 4; ++j) {
        const int col = cbase + j * 16 + lr;
        if (col < N) {
          float v = (j == 0) ? c0[i] : (j == 1) ? c1[i] : (j == 2) ? c2[i] : c3[i];
          if (bias) v += bias[col];
          if (res)  v += res[(size_t)row * res_stride + col];
          C[ro + col] = v;
        }
      }
    }
  }
}

// ---------------------------------------------------------------------------
// LSTM pointwise: gates -> (c, h), h also written to lstm_out[:, t, :]
// ---------------------------------------------------------------------------
__global__ void lstm_pointwise(const float* __restrict__ g,
                               float* __restrict__ c, float* __restrict__ h,
                               float* __restrict__ out_t) {
  const int idx = blockIdx.x * blockDim.x + threadIdx.x;
  if (idx >= B_ * H_) return;
  const int b = idx >> 9;
  const int j = idx & (H_ - 1);
  const float* gb = g + (size_t)b * G_;
  const float i_ = sigmoidf_(gb[j]);
  const float f_ = sigmoidf_(gb[H_ + j]);
  const float gg = tanhf(gb[2 * H_ + j]);
  const float o_ = sigmoidf_(gb[3 * H_ + j]);
  const float cn = f_ * c[idx] + i_ * gg;
  c[idx] = cn;
  const float hn = o_ * tanhf(cn);
  h[idx] = hn;
  out_t[(size_t)b * ((size_t)T_ * H_) + j] = hn;
}

__global__ void zero_f32(float* __restrict__ p, int n) {
  const int i = blockIdx.x * blockDim.x + threadIdx.x;
  if (i < n) p[i] = 0.0f;
}

// ---------------------------------------------------------------------------
// Attention core: per block = one (b, head, 16-row t-tile).
// kv buffer layout: [B*S, 1024], cols 0..511 = K-proj, 512..1023 = V-proj.
// ---------------------------------------------------------------------------
__global__ void attention_core(const float* __restrict__ q,
                               const float* __restrict__ kv,
                               float* __restrict__ ctx) {
  __shared__ float sc[16][S_];
  __shared__ float qs[16][HD_];
  __shared__ float red[16][16];
  __shared__ float rowmax[16], rowsum[16];

  const int blk = blockIdx.x;
  const int ttile = blk & 7;
  const int hh = (blk >> 3) & 7;
  const int b = blk >> 6;

  const int tid = threadIdx.x;
  const int r = tid >> 4;
  const int cg = tid & 15;
  const int t = ttile * 16 + r;
  const bool valid = (t < T_);

  {
    const float* qp = q + (size_t)(b * T_ + (valid ? t : 0)) * H_ + hh * HD_;
#pragma unroll
    for (int u = 0; u < 4; ++u) qs[r][cg * 4 + u] = valid ? qp[cg * 4 + u] : 0.0f;
  }
  __syncthreads();

  const float* kbase = kv + (size_t)(b * S_) * 1024 + hh * HD_;
  float lmax = -1e30f;
  for (int s = cg * 16; s < cg * 16 + 16; ++s) {
    const float* krow = kbase + (size_t)s * 1024;
    float acc = 0.0f;
#pragma unroll
    for (int d = 0; d < HD_; ++d) acc += qs[r][d] * krow[d];
    acc *= 0.125f;
    sc[r][s] = acc;
    lmax = fmaxf(lmax, acc);
  }
  red[r][cg] = lmax;
  __syncthreads();
  if (cg == 0) {
    float m = red[r][0];
    for (int u = 1; u < 16; ++u) m = fmaxf(m, red[r][u]);
    rowmax[r] = m;
  }
  __syncthreads();
  const float m = rowmax[r];
  float lsum = 0.0f;
  for (int s = cg * 16; s < cg * 16 + 16; ++s) {
    const float p = __expf(sc[r][s] - m);
    sc[r][s] = p;
    lsum += p;
  }
  red[r][cg] = lsum;
  __syncthreads();
  if (cg == 0) {
    float su = red[r][0];
    for (int u = 1; u < 16; ++u) su += red[r][u];
    rowsum[r] = su;
  }
  __syncthreads();
  const float inv = 1.0f / rowsum[r];

  const float* vbase = kv + (size_t)(b * S_) * 1024 + 512 + hh * HD_ + cg * 4;
  float a0 = 0, a1 = 0, a2 = 0, a3 = 0;
  for (int s = 0; s < S_; ++s) {
    const float p = sc[r][s];
    const float* vrow = vbase + (size_t)s * 1024;
    a0 += p * vrow[0]; a1 += p * vrow[1]; a2 += p * vrow[2]; a3 += p * vrow[3];
  }
  if (valid) {
    float* cp = ctx + (size_t)(b * T_ + t) * H_ + hh * HD_ + cg * 4;
    cp[0] = a0 * inv; cp[1] = a1 * inv; cp[2] = a2 * inv; cp[3] = a3 * inv;
  }
}

// ---------------------------------------------------------------------------
static inline void launch_gemm(const float* A, const float* W, const float* bias,
                               const float* res, long res_stride, float* C,
                               int M, int N, int K, hipStream_t stream) {
  const int nb128 = (N + 127) / 128;
  const int blocks = ((M + 63) / 64) * nb128;
  gemm_tiled<false><<<blocks, 256, 0, stream>>>(nullptr, nullptr, A, W, bias,
                                                res, res_stride, C, M, N, K, nb128);
}

extern "C" void kernel_launch(void* const* d_in, const int* in_sizes, int n_in,
                              void* d_out, int out_size, void* d_ws, size_t ws_size,
                              hipStream_t stream) {
  const int*   targets = (const int*)d_in[0];
  const float* enc     = (const float*)d_in[1];
  const float* emb     = (const float*)d_in[2];
  const float* w_ih    = (const float*)d_in[3];
  const float* w_hh    = (const float*)d_in[4];
  const float* b_ih    = (const float*)d_in[5];
  const float* b_hh    = (const float*)d_in[6];
  const float* inw     = (const float*)d_in[7];
  const float* inb     = (const float*)d_in[8];
  const float* outw    = (const float*)d_in[9];
  const float* outb    = (const float*)d_in[10];
  const float* fcw     = (const float*)d_in[11];
  const float* fcb     = (const float*)d_in[12];
  float* out = (float*)d_out;

  // Workspace carve-up (~92 MB)
  float* p = (float*)d_ws;
  float* xg       = p; p += (size_t)B_ * T_ * G_;   // [B*T, 2048]
  float* lstm_out = p; p += (size_t)B_ * T_ * H_;   // [B*T, 512]
  float* gbuf     = p; p += (size_t)B_ * G_;        // [32, 2048]
  float* hbuf     = p; p += (size_t)B_ * H_;        // [32, 512]
  float* cbuf     = p; p += (size_t)B_ * H_;        // [32, 512]
  float* qbuf     = p; p += (size_t)B_ * T_ * H_;   // [B*T, 512]
  float* kvbuf    = p; p += (size_t)B_ * S_ * 1024; // [B*S, 1024] (K|V)
  float* ctxbuf   = p; p += (size_t)B_ * T_ * H_;   // [B*T, 512]
  float* combined = qbuf;                           // q dead after attention

  // 0) zero LSTM state (hbuf, cbuf contiguous)
  zero_f32<<<(2 * B_ * H_ + 255) / 256, 256, 0, stream>>>(hbuf, 2 * B_ * H_);

  // 1) embedding gather + input-gate GEMM: bias = b_ih, res = b_hh (stride 0)
  {
    const int nb128 = G_ / 128;
    const int blocks = ((B_ * T_ + 63) / 64) * nb128;
    gemm_tiled<true><<<blocks, 256, 0, stream>>>(targets, emb, nullptr, w_ih,
                                                 b_ih, b_hh, 0, xg,
                                                 B_ * T_, G_, H_, nb128);
  }

  // 2) sequential LSTM: gates GEMM (32x2048x512, xg slice fused as residual)
  for (int t = 0; t < T_; ++t) {
    launch_gemm(hbuf, w_hh, nullptr, xg + (size_t)t * G_, (long)T_ * G_, gbuf,
                B_, G_, H_, stream);
    lstm_pointwise<<<(B_ * H_ + 255) / 256, 256, 0, stream>>>(
        gbuf, cbuf, hbuf, lstm_out + (size_t)t * H_);
  }

  // 3) projections: q from lstm_out; k|v fused (in_proj rows 512..1535)
  launch_gemm(lstm_out, inw, inb, nullptr, 0, qbuf, B_ * T_, H_, H_, stream);
  launch_gemm(enc, inw + (size_t)H_ * H_, inb + H_, nullptr, 0, kvbuf,
              B_ * S_, 2 * H_, H_, stream);

  // 4) attention core (softmax over S)
  attention_core<<<B_ * NH_ * 8, 256, 0, stream>>>(qbuf, kvbuf, ctxbuf);

  // 5) out-projection with fused residual (combined = lstm_out + attn_out)
  launch_gemm(ctxbuf, outw, outb, lstm_out, (long)H_, combined,
              B_ * T_, H_, H_, stream);

  // 6) FC logits straight into d_out: 4064 x 8000 x 512
  launch_gemm(combined, fcw, fcb, nullptr, 0, out, B_ * T_, V_, H_, stream);
}